// RealStateSpaceModel_46188078301337
// MI455X (gfx1250) — compile-verified
//
#include <hip/hip_runtime.h>

typedef __attribute__((ext_vector_type(2))) float v2f;
typedef __attribute__((ext_vector_type(8))) float v8f;

#define BATCH    8
#define LENGTH   4096
#define IN_DIM   512
#define STATE    256
#define OUT_DIM  512
#define M_TOTAL  (BATCH * LENGTH)   /* 32768 rows */
#define CHUNK    64
#define NCHUNK   (LENGTH / CHUNK)   /* 64 chunks per batch */

#define WMMA_F32(a, b, c) \
  __builtin_amdgcn_wmma_f32_16x16x4_f32(false, (a), false, (b), (short)0, (c), false, false)

// ---------------------------------------------------------------------------
// Small transpose: dst[c, r] = src[r, c].  Only used on the 0.5 MB weights.
// ---------------------------------------------------------------------------
__global__ __launch_bounds__(256) void ssm_transpose(
    const float* __restrict__ src, float* __restrict__ dst, int R, int C) {
  int idx = blockIdx.x * 256 + threadIdx.x;
  if (idx < R * C) {
    int r = idx / C, c = idx % C;
    dst[(size_t)c * R + r] = src[idx];
  }
}

// ---------------------------------------------------------------------------
// GEMM1: X[m, n] = sum_k u[m, k] * B[k, n]   (M=32768, K=512, N=256)
// Weights pre-transposed: Bt[n, k].  One block = 4 waves; each wave owns a
// 32(M) x 64(N) strip: 8 accumulators, all fragments loaded as b64.
// ---------------------------------------------------------------------------
__global__ __launch_bounds__(128) void ssm_gemm_uB(
    const float* __restrict__ u, const float* __restrict__ Bt,
    float* __restrict__ X) {
  const int lane  = threadIdx.x & 31;
  const int wave  = threadIdx.x >> 5;          // 0..3 -> N quadrant
  const int row0  = blockIdx.x * 32;           // 32-row strip of M
  const int colB  = wave * 64;
  const int mrow  = lane & 15;
  const int khalf = lane >> 4;                 // 0: K={0,1}  1: K={2,3}

  v8f c00 = {}, c01 = {}, c02 = {}, c03 = {};
  v8f c10 = {}, c11 = {}, c12 = {}, c13 = {};

  const float* arow0 = u + (size_t)(row0 + mrow) * IN_DIM + khalf * 2;
  const float* arow1 = arow0 + (size_t)16 * IN_DIM;
  const float* brow  = Bt + (size_t)(colB + mrow) * IN_DIM + khalf * 2;

  for (int k = 0; k < IN_DIM; k += 4) {
    v2f a0 = *(const v2f*)(arow0 + k);
    v2f a1 = *(const v2f*)(arow1 + k);
    v2f b0 = *(const v2f*)(brow + k);
    v2f b1 = *(const v2f*)(brow + 16 * IN_DIM + k);
    v2f b2 = *(const v2f*)(brow + 32 * IN_DIM + k);
    v2f b3 = *(const v2f*)(brow + 48 * IN_DIM + k);
    c00 = WMMA_F32(a0, b0, c00);
    c01 = WMMA_F32(a0, b1, c01);
    c02 = WMMA_F32(a0, b2, c02);
    c03 = WMMA_F32(a0, b3, c03);
    c10 = WMMA_F32(a1, b0, c10);
    c11 = WMMA_F32(a1, b1, c11);
    c12 = WMMA_F32(a1, b2, c12);
    c13 = WMMA_F32(a1, b3, c13);
  }

  // D layout: reg i -> M = i + 8*khalf, N = lane%16.
  const int srow = row0 + khalf * 8;
  for (int i = 0; i < 8; ++i) {
    float* o0 = X + (size_t)(srow + i) * STATE + colB + mrow;
    o0[0]  = c00[i];  o0[16] = c01[i];  o0[32] = c02[i];  o0[48] = c03[i];
    float* o1 = o0 + (size_t)16 * STATE;
    o1[0]  = c10[i];  o1[16] = c11[i];  o1[32] = c12[i];  o1[48] = c13[i];
  }
}

// ---------------------------------------------------------------------------
// Local scan over a 64-step chunk with zero initial state (in place on X),
// emits last value as chunk carry. block = (b, chunk), thread = state index n.
// ---------------------------------------------------------------------------
__global__ __launch_bounds__(256) void ssm_scan_local(
    const float* __restrict__ A, float* __restrict__ X,
    float* __restrict__ carries) {
  const int n  = threadIdx.x;
  const int bc = blockIdx.x;                  // b*NCHUNK + chunk
  const float a = A[n];
  size_t base = (size_t)bc * CHUNK * STATE + n;
  float v = 0.0f;
#pragma unroll 4
  for (int ti = 0; ti < CHUNK; ++ti) {
    v = fmaf(a, v, X[base + (size_t)ti * STATE]);
    X[base + (size_t)ti * STATE] = v;
  }
  carries[(size_t)bc * STATE + n] = v;
}

// ---------------------------------------------------------------------------
// Scan over chunk carries: P[b, k] = state entering chunk k (P[b,0] = x0).
// P[b,k+1] = A^64 * P[b,k] + carry[b,k].  One block per batch.
// ---------------------------------------------------------------------------
__global__ __launch_bounds__(256) void ssm_scan_carry(
    const float* __restrict__ A, const float* __restrict__ x0,
    const float* __restrict__ carries, float* __restrict__ P) {
  const int n = threadIdx.x;
  const int b = blockIdx.x;
  const float a = A[n];
  const float a2 = a * a, a4 = a2 * a2, a8 = a4 * a4;
  const float a16 = a8 * a8, a32 = a16 * a16, a64 = a32 * a32;
  float p = x0[b * STATE + n];
  for (int k = 0; k < NCHUNK; ++k) {
    const size_t idx = ((size_t)b * NCHUNK + k) * STATE + n;
    P[idx] = p;
    p = fmaf(a64, p, carries[idx]);
  }
}

// ---------------------------------------------------------------------------
// Fixup: x_global[t0+ti] = x_local[ti] + A^(ti+1) * P[b,chunk]  (in place),
// also writes final state x[:, L-1] to the tuple's second output slot.
// ---------------------------------------------------------------------------
__global__ __launch_bounds__(256) void ssm_fixup(
    const float* __restrict__ A, const float* __restrict__ P,
    float* __restrict__ X, float* __restrict__ xlast) {
  const int n     = threadIdx.x;
  const int bc    = blockIdx.x;
  const int chunk = bc & (NCHUNK - 1);
  const int b     = bc >> 6;
  const float a   = A[n];
  const float p   = P[(size_t)bc * STATE + n];
  float w = a;                                 // A^(ti+1)
  size_t base = (size_t)bc * CHUNK * STATE + n;
  float v = 0.0f;
#pragma unroll 4
  for (int ti = 0; ti < CHUNK; ++ti) {
    v = X[base + (size_t)ti * STATE] + w * p;
    X[base + (size_t)ti * STATE] = v;
    w *= a;
  }
  if (chunk == NCHUNK - 1) xlast[b * STATE + n] = v;
}

// ---------------------------------------------------------------------------
// GEMM2: Y[m, n] = sum_k X[m, k] * C[k, n]   (M=32768, K=256, N=512)
// Weights pre-transposed: Ct[n, k].  One block = 8 waves; each wave owns a
// 32(M) x 64(N) strip.
// ---------------------------------------------------------------------------
__global__ __launch_bounds__(256) void ssm_gemm_y(
    const float* __restrict__ X, const float* __restrict__ Ct,
    float* __restrict__ Y) {
  const int lane  = threadIdx.x & 31;
  const int wave  = threadIdx.x >> 5;          // 0..7 -> N strip
  const int row0  = blockIdx.x * 32;
  const int colB  = wave * 64;
  const int mrow  = lane & 15;
  const int khalf = lane >> 4;

  v8f c00 = {}, c01 = {}, c02 = {}, c03 = {};
  v8f c10 = {}, c11 = {}, c12 = {}, c13 = {};

  const float* arow0 = X + (size_t)(row0 + mrow) * STATE + khalf * 2;
  const float* arow1 = arow0 + (size_t)16 * STATE;
  const float* brow  = Ct + (size_t)(colB + mrow) * STATE + khalf * 2;

  for (int k = 0; k < STATE; k += 4) {
    v2f a0 = *(const v2f*)(arow0 + k);
    v2f a1 = *(const v2f*)(arow1 + k);
    v2f b0 = *(const v2f*)(brow + k);
    v2f b1 = *(const v2f*)(brow + 16 * STATE + k);
    v2f b2 = *(const v2f*)(brow + 32 * STATE + k);
    v2f b3 = *(const v2f*)(brow + 48 * STATE + k);
    c00 = WMMA_F32(a0, b0, c00);
    c01 = WMMA_F32(a0, b1, c01);
    c02 = WMMA_F32(a0, b2, c02);
    c03 = WMMA_F32(a0, b3, c03);
    c10 = WMMA_F32(a1, b0, c10);
    c11 = WMMA_F32(a1, b1, c11);
    c12 = WMMA_F32(a1, b2, c12);
    c13 = WMMA_F32(a1, b3, c13);
  }

  const int srow = row0 + khalf * 8;
  for (int i = 0; i < 8; ++i) {
    float* o0 = Y + (size_t)(srow + i) * OUT_DIM + colB + mrow;
    o0[0]  = c00[i];  o0[16] = c01[i];  o0[32] = c02[i];  o0[48] = c03[i];
    float* o1 = o0 + (size_t)16 * OUT_DIM;
    o1[0]  = c10[i];  o1[16] = c11[i];  o1[32] = c12[i];  o1[48] = c13[i];
  }
}

extern "C" void kernel_launch(void* const* d_in, const int* in_sizes, int n_in,
                              void* d_out, int out_size, void* d_ws, size_t ws_size,
                              hipStream_t stream) {
  const float* u  = (const float*)d_in[0];   // [8, 4096, 512]
  const float* x0 = (const float*)d_in[1];   // [8, 256]
  const float* A  = (const float*)d_in[2];   // [256]
  const float* Bm = (const float*)d_in[3];   // [512, 256]
  const float* Cm = (const float*)d_in[4];   // [256, 512]

  float* y     = (float*)d_out;                          // [8, 4096, 512]
  float* xlast = y + (size_t)M_TOTAL * OUT_DIM;          // [8, 256]

  float* X       = (float*)d_ws;                         // [32768, 256]
  float* carries = X + (size_t)M_TOTAL * STATE;          // [8, 64, 256]
  float* P       = carries + (size_t)BATCH * NCHUNK * STATE;
  float* Bt      = P + (size_t)BATCH * NCHUNK * STATE;   // [256, 512]
  float* Ct      = Bt + (size_t)IN_DIM * STATE;          // [512, 256]

  ssm_transpose <<<(IN_DIM * STATE + 255) / 256, 256, 0, stream>>>(Bm, Bt, IN_DIM, STATE);
  ssm_transpose <<<(STATE * OUT_DIM + 255) / 256, 256, 0, stream>>>(Cm, Ct, STATE, OUT_DIM);
  ssm_gemm_uB   <<<M_TOTAL / 32, 128, 0, stream>>>(u, Bt, X);
  ssm_scan_local<<<BATCH * NCHUNK, STATE, 0, stream>>>(A, X, carries);
  ssm_scan_carry<<<BATCH, STATE, 0, stream>>>(A, x0, carries, P);
  ssm_fixup     <<<BATCH * NCHUNK, STATE, 0, stream>>>(A, P, X, xlast);
  ssm_gemm_y    <<<M_TOTAL / 32, 256, 0, stream>>>(X, Ct, y);
}